// MlpAttention_45079976739230
// MI455X (gfx1250) — compile-verified
//
#include <hip/hip_runtime.h>
#include <hip/hip_bf16.h>
#include <math.h>

// ---------------- problem constants ----------------
#define NUM_CLASS 5000
#define INPUT_DIM 1024
#define EMB 512
#define ATT 512
#define CELL 1024
#define OUT 512
#define TT 800
#define BB 16
#define UU 100

typedef __attribute__((ext_vector_type(2))) float v2f;
typedef __attribute__((ext_vector_type(8))) float v8f;

__device__ __forceinline__ float sigf(float x) { return 1.0f / (1.0f + __expf(-x)); }

// =====================================================================
// Generic WMMA f32 GEMM, NT form:  C[m,n] = sum_k A[m,k] * W[n,k] (+bias[n]) (+C)
// Uses V_WMMA_F32_16X16X4_F32.  One wave computes a 16(M) x 64(N) strip.
// A VGPR layout (16x4 f32): lane l holds A[l%16][2*(l/16)+j] in elem j.
// B VGPR layout (4x16 f32): lane l holds B[2*(l/16)+j][l%16] = W[l%16][2*(l/16)+j].
// C layout: vgpr r, lane l -> C[r + 8*(l/16)][l%16].
// grid.x = M/16 tiles, grid.y * 8 waves = N/64 strips, block = 256 threads.
// =====================================================================
__global__ __launch_bounds__(256)
void wmma_gemm_nt(const float* __restrict__ A, int lda,
                  const float* __restrict__ W, int ldw,
                  float* __restrict__ C, int ldc,
                  int K, const float* __restrict__ bias, int accum) {
  const int wave = threadIdx.x >> 5;
  const int lane = threadIdx.x & 31;
  const int half = lane >> 4;
  const int lr   = lane & 15;
  const int m0   = blockIdx.x * 16;
  const int n0   = (blockIdx.y * 8 + wave) * 64;

  const float* Arow = A + (size_t)(m0 + lr) * lda + 2 * half;
  const float* Wrow0 = W + (size_t)(n0 + lr) * ldw + 2 * half;

  v8f acc0, acc1, acc2, acc3;
  if (accum) {
    #pragma unroll
    for (int r = 0; r < 8; ++r) {
      size_t row = (size_t)(m0 + r + 8 * half) * ldc + n0 + lr;
      acc0[r] = C[row];
      acc1[r] = C[row + 16];
      acc2[r] = C[row + 32];
      acc3[r] = C[row + 48];
    }
  } else {
    acc0 = (v8f){}; acc1 = (v8f){}; acc2 = (v8f){}; acc3 = (v8f){};
  }

  #pragma unroll 4
  for (int k0 = 0; k0 < K; k0 += 4) {
    v2f a2 = *(const v2f*)(Arow + k0);
    v2f b0 = *(const v2f*)(Wrow0 + k0);
    v2f b1 = *(const v2f*)(Wrow0 + (size_t)16 * ldw + k0);
    v2f b2 = *(const v2f*)(Wrow0 + (size_t)32 * ldw + k0);
    v2f b3 = *(const v2f*)(Wrow0 + (size_t)48 * ldw + k0);
    acc0 = __builtin_amdgcn_wmma_f32_16x16x4_f32(false, a2, false, b0, (short)0, acc0, false, false);
    acc1 = __builtin_amdgcn_wmma_f32_16x16x4_f32(false, a2, false, b1, (short)0, acc1, false, false);
    acc2 = __builtin_amdgcn_wmma_f32_16x16x4_f32(false, a2, false, b2, (short)0, acc2, false, false);
    acc3 = __builtin_amdgcn_wmma_f32_16x16x4_f32(false, a2, false, b3, (short)0, acc3, false, false);
  }

  float bv0 = 0.f, bv1 = 0.f, bv2 = 0.f, bv3 = 0.f;
  if (bias) {
    bv0 = bias[n0 + lr];      bv1 = bias[n0 + 16 + lr];
    bv2 = bias[n0 + 32 + lr]; bv3 = bias[n0 + 48 + lr];
  }
  #pragma unroll
  for (int r = 0; r < 8; ++r) {
    size_t row = (size_t)(m0 + r + 8 * half) * ldc + n0 + lr;
    C[row]      = acc0[r] + bv0;
    C[row + 16] = acc1[r] + bv1;
    C[row + 32] = acc2[r] + bv2;
    C[row + 48] = acc3[r] + bv3;
  }
}

// NN form (one-shot weight fusion):  C[m,n] = sum_k A[m,k] * Bm[k,n]
__global__ __launch_bounds__(256)
void wmma_gemm_nn(const float* __restrict__ A, int lda,
                  const float* __restrict__ Bm, int ldb,
                  float* __restrict__ C, int ldc, int K) {
  const int wave = threadIdx.x >> 5;
  const int lane = threadIdx.x & 31;
  const int half = lane >> 4;
  const int lr   = lane & 15;
  const int m0   = blockIdx.x * 16;
  const int n0   = (blockIdx.y * 8 + wave) * 64;

  const float* Arow = A + (size_t)(m0 + lr) * lda + 2 * half;
  v8f acc[4]; acc[0] = (v8f){}; acc[1] = (v8f){}; acc[2] = (v8f){}; acc[3] = (v8f){};

  for (int k0 = 0; k0 < K; k0 += 4) {
    v2f a2 = *(const v2f*)(Arow + k0);
    #pragma unroll
    for (int tt = 0; tt < 4; ++tt) {
      int n = n0 + tt * 16 + lr;
      v2f b2;
      b2.x = Bm[(size_t)(k0 + 2 * half) * ldb + n];
      b2.y = Bm[(size_t)(k0 + 2 * half + 1) * ldb + n];
      acc[tt] = __builtin_amdgcn_wmma_f32_16x16x4_f32(false, a2, false, b2, (short)0, acc[tt], false, false);
    }
  }
  #pragma unroll
  for (int tt = 0; tt < 4; ++tt) {
    int n = n0 + tt * 16 + lr;
    #pragma unroll
    for (int r = 0; r < 8; ++r)
      C[(size_t)(m0 + r + 8 * half) * ldc + n] = acc[tt][r];
  }
}

// =====================================================================
// Embedding with SOS shift:  labels[0]=NUM_CLASS-2, labels[u]=att_label[u-1]
// =====================================================================
__global__ void embed_kernel(const int* __restrict__ att_label,
                             const float* __restrict__ table,
                             float* __restrict__ out) {
  int u = blockIdx.x, b = blockIdx.y;
  int lab = (u == 0) ? (NUM_CLASS - 2) : att_label[(u - 1) * BB + b];
  const float* src = table + (size_t)lab * EMB;
  float* dst = out + ((size_t)u * BB + b) * EMB;
  for (int e = threadIdx.x; e < EMB; e += blockDim.x) dst[e] = src[e];
}

// =====================================================================
// LSTM cell elementwise:  gates [B,4096] split i|f|g|o
// =====================================================================
__global__ void cell_kernel(const float* __restrict__ gates,
                            float* __restrict__ c, float* __restrict__ h) {
  int idx = blockIdx.x * blockDim.x + threadIdx.x;  // 0..16383
  int b = idx >> 10, n = idx & 1023;
  const float* g = gates + (size_t)b * 4 * CELL;
  float ig = sigf(g[n]);
  float fg = sigf(g[CELL + n]);
  float gg = tanhf(g[2 * CELL + n]);
  float og = sigf(g[3 * CELL + n]);
  float cv = fg * c[idx] + ig * gg;
  cv = fminf(1.0f, fmaxf(-1.0f, cv));
  c[idx] = cv;
  h[idx] = og * tanhf(cv);
}

// =====================================================================
// alpha pass: location conv (21-tap + 8->512 expand) + base + first score.
// block = 512 threads = one (t,b); grid = (T, B).
// =====================================================================
__global__ __launch_bounds__(512)
void alpha_kernel(const float* __restrict__ att_h, const float* __restrict__ state,
                  const float* __restrict__ lss,
                  const float* __restrict__ conv1_w, const float* __restrict__ conv1_b,
                  const float* __restrict__ conv2_w, const float* __restrict__ conv2_b,
                  const float* __restrict__ w_att_v, const float* __restrict__ rnn_mask,
                  float* __restrict__ base_out, float* __restrict__ alpha_out) {
  int t = blockIdx.x, b = blockIdx.y, a = threadIdx.x;
  __shared__ float yloc[8];
  __shared__ float red[16];
  if (a < 8) {
    float acc = conv1_b[a];
    const float* lrow = lss + (size_t)b * TT;
    #pragma unroll
    for (int k = 0; k < 21; ++k) {
      int tt = t + k - 10;
      if (tt >= 0 && tt < TT) acc += conv1_w[a * 21 + k] * lrow[tt];
    }
    yloc[a] = acc;
  }
  __syncthreads();
  float floc = conv2_b[a];
  #pragma unroll
  for (int c = 0; c < 8; ++c) floc += conv2_w[a * 8 + c] * yloc[c];
  size_t idx = ((size_t)t * BB + b) * ATT + a;
  float bse = state[b * ATT + a] + att_h[idx] + floc;
  base_out[idx] = bse;
  float v = w_att_v[a] * tanhf(bse);
  #pragma unroll
  for (int off = 16; off > 0; off >>= 1) v += __shfl_xor(v, off, 32);
  if ((a & 31) == 0) red[a >> 5] = v;
  __syncthreads();
  if (a == 0) {
    float s = 0.f;
    #pragma unroll
    for (int w = 0; w < 16; ++w) s += red[w];
    s += (rnn_mask[t * BB + b] - 1.0f) * 1e10f;
    alpha_out[t * BB + b] = sigf(s);
  }
}

// =====================================================================
// shifted prefix scan of alpha * data_m over T (per (b, channel))
// =====================================================================
__global__ __launch_bounds__(512)
void memscan_kernel(const float* __restrict__ data_m, const float* __restrict__ alpha,
                    float* __restrict__ mem) {
  int b = blockIdx.x, a = threadIdx.x;
  float acc = 0.f;
  for (int t = 0; t < TT; ++t) {
    size_t idx = ((size_t)t * BB + b) * ATT + a;
    mem[idx] = acc;                       // exclusive (shifted) prefix
    acc += alpha[t * BB + b] * data_m[idx];
  }
}

// =====================================================================
// beta pass: second score with memory term; updates lss, writes alps[u].
// =====================================================================
__global__ __launch_bounds__(512)
void beta_kernel(const float* __restrict__ base, const float* __restrict__ mem,
                 const float* __restrict__ w_att_v, const float* __restrict__ rnn_mask,
                 float* __restrict__ lss, float* __restrict__ beta_out,
                 float* __restrict__ alps_u) {
  int t = blockIdx.x, b = blockIdx.y, a = threadIdx.x;
  __shared__ float red[16];
  size_t idx = ((size_t)t * BB + b) * ATT + a;
  float v = w_att_v[a] * tanhf(base[idx] + mem[idx]);
  #pragma unroll
  for (int off = 16; off > 0; off >>= 1) v += __shfl_xor(v, off, 32);
  if ((a & 31) == 0) red[a >> 5] = v;
  __syncthreads();
  if (a == 0) {
    float s = 0.f;
    #pragma unroll
    for (int w = 0; w < 16; ++w) s += red[w];
    s += (rnn_mask[t * BB + b] - 1.0f) * 1e10f;
    float bt = sigf(s);
    lss[(size_t)b * TT + t] += bt;
    beta_out[t * BB + b] = bt;
    alps_u[(size_t)b * TT + t] = bt;   // alps[u, b, t]
  }
}

// context: ctx[b,d] = sum_t data[t,b,d] * beta[t,b], masked
__global__ __launch_bounds__(256)
void ctx_kernel(const float* __restrict__ data, const float* __restrict__ beta,
                const float* __restrict__ att_mask_u, float* __restrict__ ctx_u) {
  int b = blockIdx.x, d = blockIdx.y * blockDim.x + threadIdx.x;
  float acc = 0.f;
  for (int t = 0; t < TT; ++t)
    acc += data[((size_t)t * BB + b) * INPUT_DIM + d] * beta[t * BB + b];
  ctx_u[(size_t)b * INPUT_DIM + d] = acc * att_mask_u[b];
}

__global__ void lstmp_kernel(const float* __restrict__ p,
                             const float* __restrict__ att_mask_u,
                             float* __restrict__ out_u) {
  int b = blockIdx.x, a = threadIdx.x;
  out_u[(size_t)b * OUT + a] = p[(size_t)b * OUT + a] * att_mask_u[b];
}

// =====================================================================
extern "C" void kernel_launch(void* const* d_in, const int* in_sizes, int n_in,
                              void* d_out, int out_size, void* d_ws, size_t ws_size,
                              hipStream_t stream) {
  const float* data      = (const float*)d_in[0];   // [T,B,1024]
  const float* att_mask  = (const float*)d_in[1];   // [U,B]
  const float* rnn_mask  = (const float*)d_in[2];   // [T,B,1]
  const int*   att_label = (const int*)  d_in[3];   // [U,B]
  const float* table     = (const float*)d_in[4];   // [5000,512]
  const float* w_ih      = (const float*)d_in[5];   // [4096,512]
  const float* w_hh      = (const float*)d_in[6];   // [4096,512]
  const float* b_lstm    = (const float*)d_in[7];   // [4096]
  const float* w_proj    = (const float*)d_in[8];   // [512,1024]
  const float* w_i2h     = (const float*)d_in[9];   // [512,1024]
  const float* w_p2s     = (const float*)d_in[10];  // [512,512]
  const float* w_m       = (const float*)d_in[11];  // [512,1024]
  const float* w_att_v   = (const float*)d_in[12];  // [1,512]
  const float* conv1_w   = (const float*)d_in[13];  // [8,1,21]
  const float* conv1_b   = (const float*)d_in[14];  // [8]
  const float* conv2_w   = (const float*)d_in[15];  // [512,8,1]
  const float* conv2_b   = (const float*)d_in[16];  // [512]

  float* out      = (float*)d_out;
  float* contexts = out;                                   // [U,B,1024]
  float* lstmps   = out + (size_t)UU * BB * INPUT_DIM;     // [U,B,512]
  float* alps     = lstmps + (size_t)UU * BB * OUT;        // [U,B,T]

  // workspace layout (floats); total ~27.7M floats (~111 MB)
  float* ws = (float*)d_ws;
  size_t off = 0;
  float* att_h  = ws + off; off += (size_t)TT * BB * ATT;   // data @ w_i2h^T
  float* data_m = ws + off; off += (size_t)TT * BB * ATT;   // data @ w_m^T
  float* base   = ws + off; off += (size_t)TT * BB * ATT;
  float* mem    = ws + off; off += (size_t)TT * BB * ATT;
  float* embed  = ws + off; off += (size_t)UU * BB * EMB;
  float* Wps    = ws + off; off += (size_t)ATT * CELL;      // w_p2s @ w_proj
  float* gates  = ws + off; off += (size_t)BB * 4 * CELL;
  float* cvec   = ws + off; off += (size_t)BB * CELL;
  float* hvec   = ws + off; off += (size_t)BB * CELL;
  float* pvec   = ws + off; off += (size_t)BB * OUT;
  float* state  = ws + off; off += (size_t)BB * ATT;
  float* lss    = ws + off; off += (size_t)BB * TT;         // [B,T]
  float* alphab = ws + off; off += (size_t)TT * BB;         // [T,B]
  float* betab  = ws + off; off += (size_t)TT * BB;         // [T,B]

  // deterministic state init (capturable)
  hipMemsetAsync(cvec, 0, (size_t)BB * CELL * sizeof(float), stream);
  hipMemsetAsync(pvec, 0, (size_t)BB * OUT * sizeof(float), stream);
  hipMemsetAsync(lss,  0, (size_t)BB * TT * sizeof(float), stream);

  // ---------------- precompute ----------------
  embed_kernel<<<dim3(UU, BB), 256, 0, stream>>>(att_label, table, embed);
  // att_h[T*B,512] = data[T*B,1024] @ w_i2h^T
  wmma_gemm_nt<<<dim3(TT * BB / 16, 1), 256, 0, stream>>>(
      data, INPUT_DIM, w_i2h, INPUT_DIM, att_h, ATT, INPUT_DIM, nullptr, 0);
  // data_m = data @ w_m^T  (hoisted memory projection)
  wmma_gemm_nt<<<dim3(TT * BB / 16, 1), 256, 0, stream>>>(
      data, INPUT_DIM, w_m, INPUT_DIM, data_m, ATT, INPUT_DIM, nullptr, 0);
  // Wps[512,1024] = w_p2s[512,512] @ w_proj[512,1024]  (state fusion)
  wmma_gemm_nn<<<dim3(ATT / 16, 2), 256, 0, stream>>>(
      w_p2s, OUT, w_proj, CELL, Wps, CELL, OUT);

  // ---------------- decoder scan ----------------
  for (int u = 0; u < UU; ++u) {
    // gates = embed_u @ w_ih^T + b  ;  gates += p @ w_hh^T
    wmma_gemm_nt<<<dim3(1, 8), 256, 0, stream>>>(
        embed + (size_t)u * BB * EMB, EMB, w_ih, EMB, gates, 4 * CELL, EMB, b_lstm, 0);
    wmma_gemm_nt<<<dim3(1, 8), 256, 0, stream>>>(
        pvec, OUT, w_hh, OUT, gates, 4 * CELL, OUT, nullptr, 1);
    cell_kernel<<<64, 256, 0, stream>>>(gates, cvec, hvec);
    // p = h @ w_proj^T ; state = h @ Wps^T  (= p @ w_p2s^T)
    wmma_gemm_nt<<<dim3(1, 1), 256, 0, stream>>>(
        hvec, CELL, w_proj, CELL, pvec, OUT, CELL, nullptr, 0);
    wmma_gemm_nt<<<dim3(1, 1), 256, 0, stream>>>(
        hvec, CELL, Wps, CELL, state, ATT, CELL, nullptr, 0);

    alpha_kernel<<<dim3(TT, BB), 512, 0, stream>>>(
        att_h, state, lss, conv1_w, conv1_b, conv2_w, conv2_b, w_att_v,
        rnn_mask, base, alphab);
    memscan_kernel<<<BB, 512, 0, stream>>>(data_m, alphab, mem);
    beta_kernel<<<dim3(TT, BB), 512, 0, stream>>>(
        base, mem, w_att_v, rnn_mask, lss, betab, alps + (size_t)u * BB * TT);
    ctx_kernel<<<dim3(BB, INPUT_DIM / 256), 256, 0, stream>>>(
        data, betab, att_mask + (size_t)u * BB, contexts + (size_t)u * BB * INPUT_DIM);
    lstmp_kernel<<<BB, 512, 0, stream>>>(
        pvec, att_mask + (size_t)u * BB, lstmps + (size_t)u * BB * OUT);
  }
}